// SoftThinkingMixer_7559142441428
// MI455X (gfx1250) — compile-verified
//
#include <hip/hip_runtime.h>

#define VOCAB   128000
#define DIM     2048
#define TOPK    50
#define KPAD    52          // 13 chunks * 4
#define NCHUNK  13
#define NBINS   4096
#define CAND_CAP 1024
#define BLOCK   256
#define NWAVE   (BLOCK / 32)

typedef __attribute__((ext_vector_type(2))) float v2f;
typedef __attribute__((ext_vector_type(8))) float v8f;

// monotonic float->uint key: larger float => larger key
__device__ __forceinline__ unsigned fkey(float f) {
    unsigned u = __float_as_uint(f);
    return (u & 0x80000000u) ? ~u : (u | 0x80000000u);
}
__device__ __forceinline__ float kinv(unsigned k) {
    unsigned u = (k & 0x80000000u) ? (k ^ 0x80000000u) : ~k;
    return __uint_as_float(u);
}

__global__ __launch_bounds__(BLOCK)
void soft_topk_mix(const float* __restrict__ logits,
                   const float* __restrict__ emb,
                   float* __restrict__ out)
{
    __shared__ unsigned hist[NBINS];
    __shared__ unsigned cand_key[CAND_CAP];
    __shared__ unsigned cand_idx[CAND_CAP];
    __shared__ float    w_s[KPAD];
    __shared__ unsigned i_s[KPAD];
    __shared__ unsigned cand_cnt;
    __shared__ unsigned thr_bin;
    __shared__ unsigned long long best;

    const int row = blockIdx.x;
    const int tid = threadIdx.x;
    const float* lrow = logits + (size_t)row * VOCAB;
    const float4* lrow4 = (const float4*)lrow;

    // ---- pass 1: histogram of monotonic keys (streams 512KB/row) ----
    for (int i = tid; i < NBINS; i += BLOCK) hist[i] = 0;
    if (tid == 0) cand_cnt = 0;
    __syncthreads();

    for (int i = tid; i < VOCAB / 4; i += BLOCK) {
        float4 v = lrow4[i];
        atomicAdd(&hist[fkey(v.x) >> 20], 1u);
        atomicAdd(&hist[fkey(v.y) >> 20], 1u);
        atomicAdd(&hist[fkey(v.z) >> 20], 1u);
        atomicAdd(&hist[fkey(v.w) >> 20], 1u);
    }
    __syncthreads();

    // ---- find bin containing the K-th largest ----
    if (tid == 0) {
        unsigned cum = 0; int b = NBINS - 1;
        for (; b >= 0; --b) { cum += hist[b]; if (cum >= TOPK) break; }
        thr_bin = (b < 0) ? 0u : (unsigned)b;
    }
    __syncthreads();
    const unsigned tb = thr_bin;

    // ---- pass 2: collect candidates >= threshold bin (L2-resident) ----
    for (int i = tid; i < VOCAB / 4; i += BLOCK) {
        float4 v = lrow4[i];
        float vv[4] = {v.x, v.y, v.z, v.w};
        #pragma unroll
        for (int j = 0; j < 4; ++j) {
            unsigned k = fkey(vv[j]);
            if ((k >> 20) >= tb) {
                unsigned p = atomicAdd(&cand_cnt, 1u);
                if (p < CAND_CAP) { cand_key[p] = k; cand_idx[p] = 4 * i + j; }
            }
        }
    }
    __syncthreads();
    const int cnt  = (cand_cnt < CAND_CAP) ? (int)cand_cnt : CAND_CAP;
    const int ksel = (TOPK < cnt) ? TOPK : cnt;

    // ---- exact top-K: repeated argmax via packed u64 LDS atomicMax ----
    for (int k = 0; k < ksel; ++k) {
        if (tid == 0) best = 0ull;
        __syncthreads();
        for (int i = tid; i < cnt; i += BLOCK) {
            unsigned long long pk =
                ((unsigned long long)cand_key[i] << 32) | (unsigned)i;
            atomicMax(&best, pk);
        }
        __syncthreads();
        if (tid == 0) {
            unsigned pos = (unsigned)(best & 0xffffffffull);
            unsigned key = (unsigned)(best >> 32);
            w_s[k] = kinv(key);           // logit, descending order
            i_s[k] = cand_idx[pos];
            cand_key[pos] = 0;            // remove
        }
        __syncthreads();
    }

    // ---- softmax over the selected logits (== renormalized top-k probs) ----
    if (tid == 0) {
        float m = (ksel > 0) ? w_s[0] : 0.f;   // sorted: [0] is the max
        float s = 0.f;
        for (int k = 0; k < ksel; ++k) { float e = __expf(w_s[k] - m); w_s[k] = e; s += e; }
        float inv = (s > 0.f) ? 1.f / s : 0.f;
        for (int k = 0; k < ksel; ++k) w_s[k] *= inv;
        for (int k = ksel; k < KPAD; ++k) { w_s[k] = 0.f; i_s[k] = 0; }
    }
    __syncthreads();

    // ---- weighted gather-sum via V_WMMA_F32_16X16X4_F32 ----
    // A[m][k] = emb[idx[k]][dbase+m] (ISA 16x4 A layout: lane half selects K pair)
    // B[k][n] = w[k] for all n  => every D column identical; lanes 0/16 extract.
    const int lane = tid & 31;            // wave32
    const int wave = tid >> 5;
    const int m    = lane & 15;
    const bool hi  = lane >= 16;
    float* orow = out + (size_t)row * DIM;

    // hoist weights + gather-row pointers out of the tile loop (registers)
    float wx[NCHUNK], wy[NCHUNK];
    const float* px[NCHUNK];
    const float* py[NCHUNK];
    #pragma unroll
    for (int ch = 0; ch < NCHUNK; ++ch) {
        const int k0 = ch * 4;
        const int ka = k0 + (hi ? 2 : 0);
        px[ch] = emb + (size_t)i_s[ka]     * DIM + m;
        py[ch] = emb + (size_t)i_s[ka + 1] * DIM + m;
        wx[ch] = w_s[k0 + (hi ? 2 : 0)];
        wy[ch] = w_s[k0 + (hi ? 3 : 1)];
    }

    // two independent tiles per iteration: two WMMA chains, 52 loads in flight
    const int NT = DIM / 16;              // 128 tiles
    for (int t = wave; t < NT / 2; t += NWAVE) {
        const int d0 = t * 16;
        const int d1 = (t + NT / 2) * 16;
        v8f c0 = {}, c1 = {};
        #pragma unroll
        for (int ch = 0; ch < NCHUNK; ++ch) {
            v2f a0, a1, b;
            a0.x = px[ch][d0];
            a0.y = py[ch][d0];
            a1.x = px[ch][d1];
            a1.y = py[ch][d1];
            b.x = wx[ch];
            b.y = wy[ch];
            c0 = __builtin_amdgcn_wmma_f32_16x16x4_f32(false, a0, false, b,
                                                       (short)0, c0, false, false);
            c1 = __builtin_amdgcn_wmma_f32_16x16x4_f32(false, a1, false, b,
                                                       (short)0, c1, false, false);
        }
        // contiguous 16B stores; D columns identical, lane0->M0..7, lane16->M8..15
        float4* o0 = (float4*)(orow + d0);
        float4* o1 = (float4*)(orow + d1);
        if (lane == 0) {
            o0[0] = make_float4(c0[0], c0[1], c0[2], c0[3]);
            o0[1] = make_float4(c0[4], c0[5], c0[6], c0[7]);
            o1[0] = make_float4(c1[0], c1[1], c1[2], c1[3]);
            o1[1] = make_float4(c1[4], c1[5], c1[6], c1[7]);
        } else if (lane == 16) {
            o0[2] = make_float4(c0[0], c0[1], c0[2], c0[3]);
            o0[3] = make_float4(c0[4], c0[5], c0[6], c0[7]);
            o1[2] = make_float4(c1[0], c1[1], c1[2], c1[3]);
            o1[3] = make_float4(c1[4], c1[5], c1[6], c1[7]);
        }
    }
}

extern "C" void kernel_launch(void* const* d_in, const int* in_sizes, int n_in,
                              void* d_out, int out_size, void* d_ws, size_t ws_size,
                              hipStream_t stream) {
    const float* logits = (const float*)d_in[0];   // [B,T,V] f32
    const float* emb    = (const float*)d_in[1];   // [V,D]  f32
    float* out = (float*)d_out;                    // [B,T,D] f32
    const int rows = out_size / DIM;               // B*T = 512
    soft_topk_mix<<<rows, BLOCK, 0, stream>>>(logits, emb, out);
}